// ContinuousPool_54296976556339
// MI455X (gfx1250) — compile-verified
//
#include <hip/hip_runtime.h>
#include <stdint.h>

// Problem constants (from reference):
//   x: (32, 96, 128, 128) fp32, pool_strength: (1,96,1,1) fp32 (=0.1)
//   10 steps of cur += s*(maxpool3x3_same(cur) - cur); then avgpool 2x2 -> (32,96,64,64)
#define N_IMG   (32 * 96)   // one workgroup per (n,c) image
#define HW      16384       // 128*128 floats = 64KB per image
#define W128    128
#define TSTEPS  10

__global__ __launch_bounds__(256) void cpool_kernel(const float* __restrict__ x,
                                                    const float* __restrict__ ps,
                                                    float* __restrict__ out) {
    extern __shared__ float smem[];
    float*  s_cur  = smem;           // 16384 floats (64KB): evolving image
    float*  s_h    = smem + HW;      // 16384 floats (64KB): row-wise hmax scratch
    float4* s_cur4 = reinterpret_cast<float4*>(s_cur);
    float4* s_h4   = reinterpret_cast<float4*>(s_h);

    const int tid = threadIdx.x;
    const int img = blockIdx.x;                 // 0..3071
    const float sc = ps[img % 96];              // per-channel pool strength
    const float NI = -__builtin_inff();

    // -------- Stage 1: async DMA 64KB image from global into LDS --------
    // 256 threads x 16 float4 each = 16384 floats. GVS mode: saddr(64b sgpr) + vaddr(32b byte offset).
    {
        const float* gin = x + (size_t)img * HW;
        #pragma unroll
        for (int k = 0; k < 16; ++k) {
            int g = tid + (k << 8);                                  // float4 group 0..4095
            unsigned ldsa = (unsigned)(uintptr_t)(s_cur + 4 * g);    // LDS byte address (low 32 bits)
            unsigned gofs = (unsigned)(g * 16);                      // global byte offset
            asm volatile("global_load_async_to_lds_b128 %0, %1, %2"
                         :: "v"(ldsa), "v"(gofs), "s"(gin)
                         : "memory");
        }
        asm volatile("s_wait_asynccnt 0" ::: "memory");
    }
    __syncthreads();

    // -------- Stage 2: 10 timesteps, fully resident in LDS --------
    // Separable 3x3 max with -inf border (matches reduce_window 'same' padding):
    //   pass 1: h[i][j] = max(cur[i][j-1..j+1])  (horizontal)
    //   pass 2: pooled  = max(h[i-1..i+1][j]); cur = fma(s, pooled-cur, cur) (in place)
    for (int t = 0; t < TSTEPS; ++t) {
        // pass 1: horizontal max (float4 vectorized; 32 groups per 128-wide row)
        #pragma unroll 4
        for (int k = 0; k < 16; ++k) {
            int g = tid + (k << 8);
            int colg = g & 31;                       // group within row
            float4 v = s_cur4[g];
            float left  = (colg == 0)  ? NI : s_cur[4 * g - 1];
            float right = (colg == 31) ? NI : s_cur[4 * g + 4];
            float4 h;
            h.x = fmaxf(fmaxf(left, v.x), v.y);
            h.y = fmaxf(fmaxf(v.x,  v.y), v.z);
            h.z = fmaxf(fmaxf(v.y,  v.z), v.w);
            h.w = fmaxf(fmaxf(v.z,  v.w), right);
            s_h4[g] = h;
        }
        __syncthreads();

        // pass 2: vertical max of hmax + leaky update, in place (only own element written)
        #pragma unroll 4
        for (int k = 0; k < 16; ++k) {
            int g = tid + (k << 8);
            int row = g >> 5;                        // 0..127
            float4 m = s_h4[g];
            float4 u, d;
            if (row == 0)   { u.x = u.y = u.z = u.w = NI; } else { u = s_h4[g - 32]; }
            if (row == 127) { d.x = d.y = d.z = d.w = NI; } else { d = s_h4[g + 32]; }
            float4 cv = s_cur4[g];
            float px = fmaxf(fmaxf(u.x, m.x), d.x);
            float py = fmaxf(fmaxf(u.y, m.y), d.y);
            float pz = fmaxf(fmaxf(u.z, m.z), d.z);
            float pw = fmaxf(fmaxf(u.w, m.w), d.w);
            cv.x = fmaf(sc, px - cv.x, cv.x);
            cv.y = fmaf(sc, py - cv.y, cv.y);
            cv.z = fmaf(sc, pz - cv.z, cv.z);
            cv.w = fmaf(sc, pw - cv.w, cv.w);
            s_cur4[g] = cv;
        }
        __syncthreads();
    }

    // -------- Stage 3: 2x2 avg pool, coalesced float4 stores --------
    // Output image 64x64 = 1024 float4 groups; 4 per thread.
    {
        float4* out4 = reinterpret_cast<float4*>(out) + (size_t)img * 1024;
        #pragma unroll
        for (int k = 0; k < 4; ++k) {
            int og   = tid + (k << 8);               // 0..1023
            int orow = og >> 4;                      // 16 groups per 64-wide out row
            int gb   = (orow * 2) * 32 + (og & 15) * 2;  // input float4 group
            float4 a  = s_cur4[gb];
            float4 b  = s_cur4[gb + 1];
            float4 c2 = s_cur4[gb + 32];
            float4 d2 = s_cur4[gb + 33];
            float4 o;
            o.x = (a.x + a.y + c2.x + c2.y) * 0.25f;
            o.y = (a.z + a.w + c2.z + c2.w) * 0.25f;
            o.z = (b.x + b.y + d2.x + d2.y) * 0.25f;
            o.w = (b.z + b.w + d2.z + d2.w) * 0.25f;
            out4[og] = o;
        }
    }
}

extern "C" void kernel_launch(void* const* d_in, const int* in_sizes, int n_in,
                              void* d_out, int out_size, void* d_ws, size_t ws_size,
                              hipStream_t stream) {
    (void)in_sizes; (void)n_in; (void)out_size; (void)d_ws; (void)ws_size;
    const float* x  = (const float*)d_in[0];      // (32,96,128,128) fp32
    const float* ps = (const float*)d_in[1];      // (1,96,1,1) fp32
    float* out = (float*)d_out;                   // (32,96,64,64) fp32

    const size_t shmem = 2u * HW * sizeof(float); // 128 KB (fits 320 KB/WGP, 2 WGs/WGP)
    cpool_kernel<<<dim3(N_IMG), dim3(256), shmem, stream>>>(x, ps, out);
}